// MyHeteroAttentionConv_59854664237674
// MI455X (gfx1250) — compile-verified
//
#include <hip/hip_runtime.h>

typedef __attribute__((ext_vector_type(2))) float v2f;
typedef __attribute__((ext_vector_type(8))) float v8f;

#define NEG_SLOPE 0.2f
#define EPS_F 1e-6f

// ---------------------------------------------------------------------------
// GEMM: Y[M,64] = X[M,K] @ W[K,64] in fp32 via V_WMMA_F32_16X16X4_F32.
// 256 threads = 8 waves; wave w handles output tile (mtile = blk*2 + w>>2,
// ntile = w&3), i.e. block covers 32 rows x 64 cols. W (K*64 fp32, <=32KB)
// staged in LDS once per block.
//
// VGPR layouts per CDNA5 ISA 7.12.2:
//   A 16x4 f32 : lanes 0-15 -> M=lane, VGPR0=K+0, VGPR1=K+1;
//                lanes 16-31 -> M=lane-16, VGPR0=K+2, VGPR1=K+3
//   B 4x16 f32 : lanes 0-15 -> N=lane, rows K+0/K+1; lanes 16-31 rows K+2/K+3
//   C/D 16x16  : VGPR v, lane L -> M = v + 8*(L>>4), N = L&15
// ---------------------------------------------------------------------------
__global__ __launch_bounds__(256) void hgat_gemm_f32(
    const float* __restrict__ X, const float* __restrict__ W,
    float* __restrict__ Y, int M, int K)
{
  __shared__ float Wlds[128 * 64];   // K <= 128, D_OUT = 64
  const int tid = threadIdx.x;
  const int total = K * 64;
  for (int i = tid; i < total; i += 256) Wlds[i] = W[i];
  __syncthreads();

  const int wave    = tid >> 5;
  const int lane    = tid & 31;
  const int mtile   = blockIdx.x * 2 + (wave >> 2);
  const int ntile   = wave & 3;
  const int rowbase = mtile * 16;
  if (rowbase >= M) return;

  const int halfsel = lane >> 4;        // 0: lanes 0-15, 1: lanes 16-31
  const int l15     = lane & 15;
  const int koff    = 2 * halfsel;      // K sub-offset for this half-wave
  const int col     = ntile * 16 + l15; // B/N column
  int row = rowbase + l15;
  if (row >= M) row = M - 1;            // clamp loads for ragged tail (EXEC stays full)

  const float* xrow = X + (size_t)row * K;
  v8f c = {};
  for (int k = 0; k < K; k += 4) {
    v2f a = *(const v2f*)(xrow + k + koff);          // A[M=row][k+koff .. k+koff+1]
    v2f b;
    b.x = Wlds[(k + koff)     * 64 + col];           // B[k+koff][col]
    b.y = Wlds[(k + koff + 1) * 64 + col];           // B[k+koff+1][col]
    c = __builtin_amdgcn_wmma_f32_16x16x4_f32(
        /*neg_a=*/false, a, /*neg_b=*/false, b,
        /*c_mod=*/(short)0, c, /*reuse_a=*/false, /*reuse_b=*/false);
  }

  float* yb = Y + (size_t)rowbase * 64 + ntile * 16;
#pragma unroll
  for (int v = 0; v < 8; ++v) {
    const int r = v + 8 * halfsel;                   // output row within tile
    if (rowbase + r < M) yb[(size_t)r * 64 + l15] = c[v];
  }
}

// ---------------------------------------------------------------------------
// out[n] = dot(Xf[n, 0:64], avec[0:64])   (per-node attention score half)
// ---------------------------------------------------------------------------
__global__ void hgat_node_score(const float* __restrict__ Xf,
                                const float* __restrict__ avec,
                                float* __restrict__ out, int N)
{
  const int n = blockIdx.x * blockDim.x + threadIdx.x;
  if (n >= N) return;
  const float* r = Xf + (size_t)n * 64;
  float acc = 0.f;
#pragma unroll
  for (int d = 0; d < 64; ++d) acc = fmaf(r[d], avec[d], acc);
  out[n] = acc;
}

// ---------------------------------------------------------------------------
// One wave32 per edge: lane 0 computes att = exp(leaky_relu(ss[src]+ts[tgt]))
// and atomically accumulates the denominator; all 32 lanes scatter-add a
// float2 slice of the 64-wide scaled source row into msg[tgt].
// ---------------------------------------------------------------------------
__global__ __launch_bounds__(256) void hgat_edge_scatter(
    const float* __restrict__ sx, const int* __restrict__ src,
    const int* __restrict__ tgt, const float* __restrict__ ss,
    const float* __restrict__ ts, float* __restrict__ msg,
    float* __restrict__ den, int E)
{
  const int  lane = threadIdx.x & 31;
  const long e    = (long)blockIdx.x * 8 + (threadIdx.x >> 5);
  if (e >= E) return;

  const int s = src[e];
  const int t = tgt[e];

  float att = 0.f;
  if (lane == 0) {
    const float score = ss[s] + ts[t];
    const float lr    = score > 0.f ? score : NEG_SLOPE * score;
    att = __expf(lr);
    atomicAdd(den + t, att);
  }
  att = __shfl(att, 0, 32);  // wave32 broadcast

  const float2 v = ((const float2*)(sx + (size_t)s * 64))[lane];
  float* mrow = msg + (size_t)t * 64 + 2 * lane;
  atomicAdd(mrow,     v.x * att);
  atomicAdd(mrow + 1, v.y * att);
}

// ---------------------------------------------------------------------------
// out = tx + msg / (den + eps)   (msg already accumulated in-place in out)
// ---------------------------------------------------------------------------
__global__ void hgat_finalize(const float* __restrict__ tx,
                              const float* __restrict__ den,
                              float* __restrict__ out, int N)
{
  const int i = blockIdx.x * blockDim.x + threadIdx.x;
  if (i >= N * 64) return;
  const int n = i >> 6;
  out[i] = tx[i] + out[i] / (den[n] + EPS_F);
}

extern "C" void kernel_launch(void* const* d_in, const int* in_sizes, int n_in,
                              void* d_out, int out_size, void* d_ws, size_t ws_size,
                              hipStream_t stream)
{
  (void)n_in; (void)ws_size;

  const float* x_user   = (const float*)d_in[0];
  const float* x_item   = (const float*)d_in[1];
  const float* W_ui_src = (const float*)d_in[2];
  const float* W_ui_tgt = (const float*)d_in[3];
  const float* W_iu_src = (const float*)d_in[4];
  const float* W_iu_tgt = (const float*)d_in[5];
  const float* a_ui     = (const float*)d_in[6];
  const float* a_iu     = (const float*)d_in[7];
  const int*   edge_ui  = (const int*)d_in[8];
  const int*   edge_iu  = (const int*)d_in[9];

  const int D_OUT = in_sizes[6] / 2;        // 64
  const int K     = in_sizes[2] / D_OUT;    // 128
  const int Nu    = in_sizes[0] / K;        // 100000
  const int Ni    = in_sizes[1] / K;        // 100000
  const int Eui   = in_sizes[8] / 2;        // 1000000
  const int Eiu   = in_sizes[9] / 2;        // 1000000

  // workspace layout (fp32)
  float* ws    = (float*)d_ws;
  float* sx_ui = ws; ws += (size_t)Nu * 64;     // x_user @ W_ui_src
  float* tx_ui = ws; ws += (size_t)Ni * 64;     // x_item @ W_ui_tgt
  float* sx_iu = ws; ws += (size_t)Ni * 64;     // x_item @ W_iu_src
  float* tx_iu = ws; ws += (size_t)Nu * 64;     // x_user @ W_iu_tgt
  float* ss_ui = ws; ws += Nu;
  float* ts_ui = ws; ws += Ni;
  float* ss_iu = ws; ws += Ni;
  float* ts_iu = ws; ws += Nu;
  float* den_user = ws;
  float* den_item = ws + Nu;

  float* out_user = (float*)d_out;                        // [Nu,64]
  float* out_item = out_user + (size_t)Nu * 64;           // [Ni,64]

  // msg accumulates in d_out; den in ws -> zero them every call
  hipMemsetAsync(d_out, 0, (size_t)out_size * sizeof(float), stream);
  hipMemsetAsync(den_user, 0, (size_t)(Nu + Ni) * sizeof(float), stream);

  // --- dense projections (WMMA fp32) ---
  const dim3 blk(256);
  const int gU = (Nu + 31) / 32, gI = (Ni + 31) / 32;
  hgat_gemm_f32<<<gU, blk, 0, stream>>>(x_user, W_ui_src, sx_ui, Nu, K);
  hgat_gemm_f32<<<gI, blk, 0, stream>>>(x_item, W_ui_tgt, tx_ui, Ni, K);
  hgat_gemm_f32<<<gI, blk, 0, stream>>>(x_item, W_iu_src, sx_iu, Ni, K);
  hgat_gemm_f32<<<gU, blk, 0, stream>>>(x_user, W_iu_tgt, tx_iu, Nu, K);

  // --- per-node score halves ---
  const int tb = 256;
  hgat_node_score<<<(Nu + tb - 1) / tb, tb, 0, stream>>>(sx_ui, a_ui,         ss_ui, Nu);
  hgat_node_score<<<(Ni + tb - 1) / tb, tb, 0, stream>>>(tx_ui, a_ui + D_OUT, ts_ui, Ni);
  hgat_node_score<<<(Ni + tb - 1) / tb, tb, 0, stream>>>(sx_iu, a_iu,         ss_iu, Ni);
  hgat_node_score<<<(Nu + tb - 1) / tb, tb, 0, stream>>>(tx_iu, a_iu + D_OUT, ts_iu, Nu);

  // --- edge softmax numerator / denominator scatter (bandwidth-bound phase) ---
  hgat_edge_scatter<<<(Eui + 7) / 8, 256, 0, stream>>>(
      sx_ui, edge_ui, edge_ui + Eui, ss_ui, ts_ui, out_item, den_item, Eui);
  hgat_edge_scatter<<<(Eiu + 7) / 8, 256, 0, stream>>>(
      sx_iu, edge_iu, edge_iu + Eiu, ss_iu, ts_iu, out_user, den_user, Eiu);

  // --- out = tx + msg/(den+eps) ---
  hgat_finalize<<<((Nu * 64) + tb - 1) / tb, tb, 0, stream>>>(tx_iu, den_user, out_user, Nu);
  hgat_finalize<<<((Ni * 64) + tb - 1) / tb, tb, 0, stream>>>(tx_ui, den_item, out_item, Ni);
}